// PoolingCBP_20005957665171
// MI455X (gfx1250) — compile-verified
//
#include <hip/hip_runtime.h>
#include <math.h>

// ---------------------------------------------------------------------------
// Compact Bilinear Pooling, fused for MI455X (gfx1250, wave32, WMMA).
//   x:(16,2048,14,14) f32, h1/h2:(2048) i32, w1/w2:(2048) f32 -> out:(16,16384) f32
//
// Kernel 1 (one WG per spatial location, 3136 WGs, 512 thr, 160 KB LDS):
//   - scatter count-sketches s1 (re) and s2 (im) into LDS with ds_add_f32
//   - one 16384-pt complex FFT in LDS; radix-16 stages are 16x16 complex
//     matmuls on v_wmma_f32_16x16x4_f32 (tcFFT style), final radix-4 VALU;
//     all twiddles from a quarter-wave LDS table, DFT16 operand registers
//     hoisted out of the tile loop
//   - unpack R1,R2 via conjugate symmetry (in digit-reversed coords),
//     pointwise product, global_atomic_add_f32 into per-batch spectrum (L2)
// Kernel 2 (one WG per batch, 16 WGs):
//   - inverse FFT (conjugate-transposed DIT stage order: r4 then 3x WMMA r16)
//     consumes permuted spectrum, emits natural order
//   - signed sqrt + L2 normalize (LDS reduction)
// ---------------------------------------------------------------------------

typedef __attribute__((ext_vector_type(2))) float v2f;
typedef __attribute__((ext_vector_type(8))) float v8f;

#define D_PROJ   16384
#define QTAB     4096      // quarter-wave table size (D_PROJ/4)
#define C_CH     2048
#define HW_SP    196
#define B_BATCH  16
#define NLOC     3136      // 16*196
#define NTHR     512       // 16 waves of 32
#define TWO_PI_F 6.28318530717958647692f

__device__ __forceinline__ v8f wmma_k4(v2f a, v2f b, v8f c) {
  // D(16x16,f32) = A(16x4,f32) * B(4x16,f32) + C
  return __builtin_amdgcn_wmma_f32_16x16x4_f32(false, a, false, b, (short)0, c,
                                               false, false);
}

// Mixed-radix (16,16,16,4) digit reversal: permuted position p -> frequency f
__device__ __forceinline__ int rho(int p) {
  return ((p >> 10) & 15) | (((p >> 6) & 15) << 4) | (((p >> 2) & 15) << 8) |
         ((p & 3) << 12);
}
__device__ __forceinline__ int rho_inv(int k) {
  return ((k & 15) << 10) | (((k >> 4) & 15) << 6) | (((k >> 8) & 15) << 2) |
         ((k >> 12) & 3);
}

// e^{+i*2*pi*t/16384} from the quarter-wave LDS table, t in [0, 16384).
// Branch-free quadrant reconstruction (v_cndmask, no EXEC games).
__device__ __forceinline__ void twget(const float* __restrict__ twc,
                                      const float* __restrict__ tws, int t,
                                      float& c, float& s) {
  const int r = t & (QTAB - 1);
  const float C = twc[r];
  const float S = tws[r];
  const int q = (t >> 12) & 3;
  const bool swap = (q & 1) != 0;
  const float cv = swap ? S : C;
  const float sv = swap ? C : S;
  c = ((q == 1) || (q == 2)) ? -cv : cv;
  s = (q >= 2) ? -sv : sv;
}

// One radix-16 stage over blocks of length L (B = L/16 columns per block).
// Forward (INV=0): DIF, DFT16^- then post-twiddle w_L^{-b*a'}.
// Inverse (INV=1): conjugate transpose: pre-twiddle w_L^{+b*a'} then DFT16^+.
// 1024 sub-sequences -> 64 16x16 complex tiles; each tile = 16 WMMAs.
template <int L, bool INV>
__device__ void r16_stage(float* re, float* im, const float* twc,
                          const float* tws, int wv, int lane) {
  constexpr int Bsz = L / 16;
  constexpr int STEP = D_PROJ / L;  // twiddle index multiplier
  const int hi = lane >> 4;         // half-wave select
  const int lj = lane & 15;         // column within tile / A row

  // DFT16 matrix operands are tile-invariant: build them once, in registers,
  // so the compiler doesn't have to reload them past the in-loop ds_stores.
  // A layout (16x4 f32): row = lane%16, K = {e, e+2 for upper half-wave}.
  v2f ArV[4], AiV[4];
#pragma unroll
  for (int kk = 0; kk < 4; ++kk) {
#pragma unroll
    for (int e = 0; e < 2; ++e) {
      const int a = 4 * kk + e + 2 * hi;
      float c16, s16;
      twget(twc, tws, ((lj * a) & 15) << 10, c16, s16);
      ArV[kk][e] = c16;
      AiV[kk][e] = INV ? s16 : -s16;  // DFT16^+ vs DFT16^-
    }
  }

  for (int g = wv; g < 64; g += 16) {
    const int col = g * 16 + lj;           // global sub-sequence index
    const int b_ = col % Bsz;
    const int S = (col / Bsz) * L;

    v8f accRR = {0.f, 0.f, 0.f, 0.f, 0.f, 0.f, 0.f, 0.f}; // Fr*Mr
    v8f accII = {0.f, 0.f, 0.f, 0.f, 0.f, 0.f, 0.f, 0.f}; // Fi*Mi
    v8f accVI = {0.f, 0.f, 0.f, 0.f, 0.f, 0.f, 0.f, 0.f}; // Fr*Mi + Fi*Mr

#pragma unroll
    for (int kk = 0; kk < 4; ++kk) {
      v2f Br, Bi;
#pragma unroll
      for (int e = 0; e < 2; ++e) {
        // B operand: data chunk (4x16): row a, column lj (this lane's col).
        const int a = 4 * kk + e + 2 * hi;
        float mr = re[S + b_ + Bsz * a];
        float mi = im[S + b_ + Bsz * a];
        if (INV) {  // pre-twiddle w_L^{+b*a'} on input digit (== a here)
          float ct, st;
          twget(twc, tws, (b_ * a) * STEP, ct, st);
          const float tr = mr * ct - mi * st;
          mi = mr * st + mi * ct;
          mr = tr;
        }
        Br[e] = mr;
        Bi[e] = mi;
      }
      accRR = wmma_k4(ArV[kk], Br, accRR);
      accII = wmma_k4(AiV[kk], Bi, accII);
      accVI = wmma_k4(ArV[kk], Bi, accVI);
      accVI = wmma_k4(AiV[kk], Br, accVI);
    }

#pragma unroll
    for (int v = 0; v < 8; ++v) {
      const int ap = v + 8 * hi;            // output digit (D rows layout)
      float Vr = accRR[v] - accII[v];
      float Vi = accVI[v];
      if (!INV) {  // post-twiddle w_L^{-b*a'} = (ct, -st)
        float ct, st;
        twget(twc, tws, (b_ * ap) * STEP, ct, st);
        const float tr = Vr * ct + Vi * st;
        Vi = -Vr * st + Vi * ct;
        Vr = tr;
      }
      re[S + b_ + Bsz * ap] = Vr;
      im[S + b_ + Bsz * ap] = Vi;
    }
  }
}

// Radix-4 stage on contiguous blocks of 4 (twiddle-free: b==0).
template <bool INV>
__device__ void r4_stage(float* re, float* im, int tid) {
  const float si = INV ? 1.0f : -1.0f;  // sign of i in DFT4
  for (int blk = tid; blk < D_PROJ / 4; blk += NTHR) {
    const int S = blk * 4;
    const float r0 = re[S], i0 = im[S], r1 = re[S + 1], i1 = im[S + 1];
    const float r2 = re[S + 2], i2 = im[S + 2], r3 = re[S + 3], i3 = im[S + 3];
    const float ar = r0 + r2, ai = i0 + i2;
    const float br = r0 - r2, bi = i0 - i2;
    const float cr = r1 + r3, ci = i1 + i3;
    const float dr = r1 - r3, di = i1 - i3;
    re[S] = ar + cr;              im[S] = ai + ci;
    re[S + 2] = ar - cr;          im[S + 2] = ai - ci;
    re[S + 1] = br - si * di;     im[S + 1] = bi + si * dr;
    re[S + 3] = br + si * di;     im[S + 3] = bi - si * dr;
  }
}

template <bool INV>
__device__ void fft16384(float* re, float* im, const float* twc,
                         const float* tws, int tid) {
  const int wv = tid >> 5, lane = tid & 31;
  if (INV) {  // permuted in -> natural out (conjugate transpose of forward)
    r4_stage<true>(re, im, tid);                        __syncthreads();
    r16_stage<64, true>(re, im, twc, tws, wv, lane);    __syncthreads();
    r16_stage<1024, true>(re, im, twc, tws, wv, lane);  __syncthreads();
    r16_stage<16384, true>(re, im, twc, tws, wv, lane); __syncthreads();
  } else {    // natural in -> digit-reversed out (DIF)
    r16_stage<16384, false>(re, im, twc, tws, wv, lane); __syncthreads();
    r16_stage<1024, false>(re, im, twc, tws, wv, lane);  __syncthreads();
    r16_stage<64, false>(re, im, twc, tws, wv, lane);    __syncthreads();
    r4_stage<false>(re, im, tid);                        __syncthreads();
  }
}

__device__ __forceinline__ void fill_twiddle(float* twc, float* tws, int tid) {
  for (int t = tid; t < QTAB; t += NTHR) {
    float s, c;
    sincosf(TWO_PI_F * (float)t * (1.0f / (float)D_PROJ), &s, &c);
    twc[t] = c;
    tws[t] = s;
  }
}

// ---------------------------------------------------------------------------
__global__ __launch_bounds__(NTHR, 1) void cbp_forward(
    const float* __restrict__ x, const float* __restrict__ w1,
    const float* __restrict__ w2, const int* __restrict__ h1,
    const int* __restrict__ h2, float* __restrict__ accR,
    float* __restrict__ accI) {
  extern __shared__ float smem[];
  float* re = smem;                   // s1 / Re(Z)
  float* im = smem + D_PROJ;          // s2 / Im(Z)
  float* twc = smem + 2 * D_PROJ;     // quarter-wave cos
  float* tws = twc + QTAB;            // quarter-wave sin
  const int tid = threadIdx.x;
  const int n = blockIdx.x;           // spatial location
  const int b = n / HW_SP;
  const int hw = n % HW_SP;

  for (int p = tid; p < D_PROJ; p += NTHR) { re[p] = 0.f; im[p] = 0.f; }
  fill_twiddle(twc, tws, tid);
  __syncthreads();

  // Count-sketch scatter: z = s1 + i*s2 built directly with LDS f32 atomics.
  // x gather is stride-784B; prefetch the next iteration's line.
  for (int c = tid; c < C_CH; c += NTHR) {
    if (c + NTHR < C_CH)
      __builtin_prefetch(&x[(b * C_CH + c + NTHR) * HW_SP + hw], 0, 0);
    const float v = x[(b * C_CH + c) * HW_SP + hw];
    atomicAdd(&re[h1[c]], w1[c] * v);
    atomicAdd(&im[h2[c]], w2[c] * v);
  }
  __syncthreads();

  fft16384<false>(re, im, twc, tws, tid); // Z = FFT(s1+i*s2), digit-reversed

  // Unpack R1=FFT(s1), R2=FFT(s2) via conjugate symmetry (in permuted
  // coords), multiply, accumulate per-batch spectrum (stays permuted).
  float* aR = accR + b * D_PROJ;
  float* aI = accI + b * D_PROJ;
  for (int p = tid; p < D_PROJ; p += NTHR) {
    const int f = rho(p);
    const int p2 = rho_inv((D_PROJ - f) & (D_PROJ - 1));
    const float Zr = re[p], Zi = im[p];
    const float Wr = re[p2], Wi = -im[p2];        // conj(Z[N-f])
    const float R1r = 0.5f * (Zr + Wr), R1i = 0.5f * (Zi + Wi);
    const float R2r = 0.5f * (Zi - Wi), R2i = -0.5f * (Zr - Wr);
    atomicAdd(&aR[p], R1r * R2r - R1i * R2i);
    atomicAdd(&aI[p], R1r * R2i + R1i * R2r);
  }
}

__global__ __launch_bounds__(NTHR, 1) void cbp_inverse(
    const float* __restrict__ accR, const float* __restrict__ accI,
    float* __restrict__ out) {
  extern __shared__ float smem[];
  float* re = smem;
  float* im = smem + D_PROJ;
  float* twc = smem + 2 * D_PROJ;
  float* tws = twc + QTAB;
  float* red = tws + QTAB;
  const int tid = threadIdx.x;
  const int b = blockIdx.x;

  for (int p = tid; p < D_PROJ; p += NTHR) {
    re[p] = accR[b * D_PROJ + p];
    im[p] = accI[b * D_PROJ + p];
  }
  fill_twiddle(twc, tws, tid);
  if (tid == 0) red[0] = 0.f;
  __syncthreads();

  fft16384<true>(re, im, twc, tws, tid);  // N * IDFT, natural order

  const float invN = 1.0f / (float)D_PROJ;
  float part = 0.f;
  for (int p = tid; p < D_PROJ; p += NTHR) {
    const float y = re[p] * invN;
    const float t = copysignf(sqrtf(fabsf(y) + 1e-8f), y);
    re[p] = t;
    part += t * t;
  }
  atomicAdd(&red[0], part);
  __syncthreads();

  const float s = 1.0f / sqrtf(red[0] + 1e-8f);
  for (int p = tid; p < D_PROJ; p += NTHR) out[b * D_PROJ + p] = re[p] * s;
}

// ---------------------------------------------------------------------------
extern "C" void kernel_launch(void* const* d_in, const int* in_sizes, int n_in,
                              void* d_out, int out_size, void* d_ws,
                              size_t ws_size, hipStream_t stream) {
  (void)in_sizes; (void)n_in; (void)out_size; (void)ws_size;
  const float* x  = (const float*)d_in[0];
  const float* w1 = (const float*)d_in[1];
  const float* w2 = (const float*)d_in[2];
  const int*   h1 = (const int*)d_in[3];
  const int*   h2 = (const int*)d_in[4];
  float* accR = (float*)d_ws;                  // 16*16384 f32
  float* accI = accR + B_BATCH * D_PROJ;       // 16*16384 f32  (2 MB total)

  const size_t lds_fwd = (2u * D_PROJ + 2u * QTAB) * sizeof(float);  // 160 KB
  const size_t lds_inv = lds_fwd + 64;                               // + red
  hipFuncSetAttribute(reinterpret_cast<const void*>(cbp_forward),
                      hipFuncAttributeMaxDynamicSharedMemorySize,
                      (int)lds_fwd);
  hipFuncSetAttribute(reinterpret_cast<const void*>(cbp_inverse),
                      hipFuncAttributeMaxDynamicSharedMemorySize,
                      (int)lds_inv);

  hipMemsetAsync(d_ws, 0, (size_t)2 * B_BATCH * D_PROJ * sizeof(float),
                 stream);
  cbp_forward<<<NLOC, NTHR, lds_fwd, stream>>>(x, w1, w2, h1, h2, accR, accI);
  cbp_inverse<<<B_BATCH, NTHR, lds_inv, stream>>>(accR, accI, (float*)d_out);
}